// PureGNN_27659589386259
// MI455X (gfx1250) — compile-verified
//
#include <hip/hip_runtime.h>
#include <hip/hip_bf16.h>

#define N_TOTAL 100000
#define DIM     128
#define NNZ     800000
#define N_USER  50000
#define N_ITEM  50000

typedef float v2f __attribute__((ext_vector_type(2)));
typedef float v8f __attribute__((ext_vector_type(8)));

// ---------------------------------------------------------------------------
// Kernel 1: zero the two SpMM accumulators living in d_ws (float4 stores).
// ---------------------------------------------------------------------------
__global__ void gnn_zero_ws(float4* __restrict__ ws, int n4) {
    int i = blockIdx.x * blockDim.x + threadIdx.x;
    if (i < n4) ws[i] = make_float4(0.f, 0.f, 0.f, 0.f);
}

// ---------------------------------------------------------------------------
// Kernel 2: COO SpMM scatter.  One wave (32 lanes) per edge; lane c handles
// 4 contiguous floats of the 128-wide row.  Gather is a 128-bit load from
// ebs (L2-resident), scatter is hardware global_atomic_add_f32 into the
// L2-resident accumulator.
// ---------------------------------------------------------------------------
__global__ void gnn_spmm_scatter(const float* __restrict__ ebs,
                                 const int*   __restrict__ rows,
                                 const int*   __restrict__ cols,
                                 const float* __restrict__ vals,
                                 float*       __restrict__ acc) {
    long long tid = (long long)blockIdx.x * blockDim.x + threadIdx.x;
    int e = (int)(tid >> 5);
    int c = ((int)tid & 31) * 4;
    if (e >= NNZ) return;

    int   r = rows[e];
    int   s = cols[e];
    float v = vals[e];

    const float4 src = *(const float4*)(ebs + (size_t)s * DIM + c);
    float* dst = acc + (size_t)r * DIM + c;
    unsafeAtomicAdd(dst + 0, v * src.x);
    unsafeAtomicAdd(dst + 1, v * src.y);
    unsafeAtomicAdd(dst + 2, v * src.z);
    unsafeAtomicAdd(dst + 3, v * src.w);
}

// ---------------------------------------------------------------------------
// Kernel 3: relu(A @ W) with V_WMMA_F32_16X16X4_F32.
// Block = 256 threads = 8 waves.  blockIdx.x picks a 16-row M-tile of A
// (50000 = 3125*16 exactly); wave w owns the 16-col N-tile w (128 = 8*16).
// K loop: 128/4 = 32 WMMA steps.
//
// Per-lane fragment layout (CDNA5 ISA 7.12.2, 32-bit A 16x4 / B 4x16 /
// C/D 16x16):
//   lane = hi*16 + li  (hi = lane>>4, li = lane&15)
//   A: row M = li,          a = { A[M][k + 2*hi], A[M][k + 2*hi + 1] }
//   B: col N = li,          b = { W[k + 2*hi][N], W[k + 2*hi + 1][N] }
//   C/D: VGPR r holds element (M = r + 8*hi, N = li)
// ---------------------------------------------------------------------------
__global__ void gnn_gemm_relu_wmma(const float* __restrict__ A,
                                   const float* __restrict__ W,
                                   float* __restrict__ out) {
    const int wave = threadIdx.x >> 5;     // 0..7 -> N tile
    const int lane = threadIdx.x & 31;
    const int hi   = lane >> 4;            // 0 or 1
    const int li   = lane & 15;

    const int row0 = blockIdx.x * 16;      // M tile base
    const int col  = wave * 16 + li;       // this lane's N within D

    const float* arow = A + (size_t)(row0 + li) * DIM;

    v8f c;
    c[0]=0.f; c[1]=0.f; c[2]=0.f; c[3]=0.f; c[4]=0.f; c[5]=0.f; c[6]=0.f; c[7]=0.f;

#pragma unroll 8
    for (int k = 0; k < DIM; k += 4) {
        const int kk = k + hi * 2;                       // even -> 8B aligned
        v2f a = *(const v2f*)(arow + kk);                // global_load_b64
        v2f b;
        b.x = W[(size_t)kk * DIM + col];
        b.y = W[(size_t)(kk + 1) * DIM + col];
        c = __builtin_amdgcn_wmma_f32_16x16x4_f32(
                /*neg_a=*/false, a, /*neg_b=*/false, b,
                /*c_mod=*/(short)0, c, /*reuse_a=*/false, /*reuse_b=*/false);
    }

#pragma unroll
    for (int r = 0; r < 8; ++r) {
        const int row = row0 + r + 8 * hi;
        float v = c[r];
        out[(size_t)row * DIM + col] = v > 0.f ? v : 0.f;
    }
}

// ---------------------------------------------------------------------------
// Launch: zero ws -> scatter(user) -> scatter(item) -> gemm(user) -> gemm(item)
// ---------------------------------------------------------------------------
extern "C" void kernel_launch(void* const* d_in, const int* in_sizes, int n_in,
                              void* d_out, int out_size, void* d_ws, size_t ws_size,
                              hipStream_t stream) {
    const float* ebs        = (const float*)d_in[0];
    const int*   rows_user  = (const int*)  d_in[1];
    const int*   cols_user  = (const int*)  d_in[2];
    const float* vals_user  = (const float*)d_in[3];
    const float* W_user     = (const float*)d_in[4];
    const int*   rows_item  = (const int*)  d_in[5];
    const int*   cols_item  = (const int*)  d_in[6];
    const float* vals_item  = (const float*)d_in[7];
    const float* W_item     = (const float*)d_in[8];

    float* acc_user = (float*)d_ws;
    float* acc_item = acc_user + (size_t)N_USER * DIM;
    float* out      = (float*)d_out;

    // 1) zero accumulators: 2 * 50000 * 128 floats = 3.2M float4
    {
        int n4 = (N_USER + N_ITEM) * DIM / 4;
        int blocks = (n4 + 255) / 256;
        gnn_zero_ws<<<blocks, 256, 0, stream>>>((float4*)d_ws, n4);
    }

    // 2) SpMM scatter: 32 lanes per edge
    {
        long long threads = (long long)NNZ * 32;
        int blocks = (int)((threads + 255) / 256);
        gnn_spmm_scatter<<<blocks, 256, 0, stream>>>(ebs, rows_user, cols_user,
                                                     vals_user, acc_user);
        gnn_spmm_scatter<<<blocks, 256, 0, stream>>>(ebs, rows_item, cols_item,
                                                     vals_item, acc_item);
    }

    // 3) GEMM + ReLU via WMMA f32 16x16x4
    {
        int blocks = N_USER / 16;   // 3125, exact
        gnn_gemm_relu_wmma<<<blocks, 256, 0, stream>>>(acc_user, W_user, out);
        gnn_gemm_relu_wmma<<<blocks, 256, 0, stream>>>(acc_item, W_item,
                                                       out + (size_t)N_USER * DIM);
    }
}